// RangeAwareL1Loss_40020505264451
// MI455X (gfx1250) — compile-verified
//
#include <hip/hip_runtime.h>
#include <math.h>

// Range-aware L1 loss, single streaming pass (HBM-roofline bound: 134 MB ~ 5.8 us).
// ws layout (floats): [0..30] per-bin valid counts, [31..61] per-bin sum|pred-target|, [62] total_valid
#define NBINS     31
#define NREP      4                       // replicated LDS sub-histograms (hot-bin conflict reduction)
#define WS_CNT    0
#define WS_SUM    NBINS
#define WS_VALID  (2 * NBINS)
#define WS_FLOATS 63

// packed LDS accumulator: count in bits [63:44], |diff| fixed-point (2^-18) in bits [43:0]
#define CNT_SHIFT 44
#define FIX_SCALE 262144.0f               // 2^18
#define FIX_INV   (1.0f / 262144.0f)
#define FIX_MASK  ((1ull << CNT_SHIFT) - 1ull)

__device__ __forceinline__ void accum_pixel(float p, float t, int rep,
                                            unsigned long long (*s_hist)[NBINS],
                                            float& vcnt) {
  if (t != -1.0f) {                       // valid (-1 sentinel)
    vcnt += 1.0f;
    float nat = __expf(t) - 1.0f;         // v_exp_f32-based expm1; binning only needs floor
    if (nat >= 0.0f) {                    // in_range
      int b = (int)fminf(floorf(nat), 30.0f);     // clip to [0,30]
      unsigned long long pk = (1ull << CNT_SHIFT)
          + (unsigned long long)__float2uint_rn(fabsf(p - t) * FIX_SCALE);
      atomicAdd(&s_hist[rep][b], pk);     // single ds_add_u64 per pixel
    }
  }
}

__global__ void rangeloss_zero_ws(float* ws) {
  int i = threadIdx.x;
  if (i < WS_FLOATS) ws[i] = 0.0f;
}

__global__ __launch_bounds__(256) void rangeloss_hist(
    const float* __restrict__ pred, const float* __restrict__ target,
    float* __restrict__ ws, int n4, int n) {
  __shared__ unsigned long long s_hist[NREP][NBINS];
  __shared__ float s_valid;
  const int t = threadIdx.x;
  {
    unsigned long long* lin = &s_hist[0][0];
    if (t < NREP * NBINS) lin[t] = 0ull;
    if (t == 0) s_valid = 0.0f;
  }
  __syncthreads();

  const int rep = t & (NREP - 1);         // lanes with equal bin spread over 4 banks-apart replicas
  float vcnt = 0.0f;
  const float4* __restrict__ p4 = (const float4*)pred;
  const float4* __restrict__ t4 = (const float4*)target;
  const int stride = (int)(gridDim.x * blockDim.x);

  for (int i = (int)(blockIdx.x * blockDim.x) + t; i < n4; i += stride) {
    if (i + stride < n4) {                // gfx1250 global_prefetch_b8, one stride ahead
      __builtin_prefetch(&p4[i + stride], 0, 0);
      __builtin_prefetch(&t4[i + stride], 0, 0);
    }
    float4 pv = p4[i];                    // global_load_b128
    float4 tv = t4[i];                    // global_load_b128
    accum_pixel(pv.x, tv.x, rep, s_hist, vcnt);
    accum_pixel(pv.y, tv.y, rep, s_hist, vcnt);
    accum_pixel(pv.z, tv.z, rep, s_hist, vcnt);
    accum_pixel(pv.w, tv.w, rep, s_hist, vcnt);
  }
  for (int i = 4 * n4 + (int)(blockIdx.x * blockDim.x) + t; i < n; i += stride) {
    accum_pixel(pred[i], target[i], rep, s_hist, vcnt);
  }

  // wave32 butterfly reduce of per-thread valid count
  #pragma unroll
  for (int off = 16; off > 0; off >>= 1)
    vcnt += __shfl_down(vcnt, off, 32);
  if ((t & 31) == 0) atomicAdd(&s_valid, vcnt);
  __syncthreads();

  // fold replicas, unpack, one global f32 atomic pair per bin per block
  if (t < NBINS) {
    unsigned long long pk = s_hist[0][t] + s_hist[1][t] + s_hist[2][t] + s_hist[3][t];
    float cnt  = (float)(pk >> CNT_SHIFT);
    float sumv = (float)(pk & FIX_MASK) * FIX_INV;
    atomicAdd(&ws[WS_CNT + t], cnt);
    atomicAdd(&ws[WS_SUM + t], sumv);
  }
  if (t == 0) atomicAdd(&ws[WS_VALID], s_valid);
}

__global__ void rangeloss_finalize(const float* __restrict__ ws,
                                   float* __restrict__ out) {
  const int t = threadIdx.x;              // one wave (32 threads)
  const float total = ws[WS_VALID];
  float part = 0.0f;
  if (t < NBINS) {
    float freq = ws[WS_CNT + t] / total;
    float w = 1.0f / (sqrtf(freq) + 1e-6f);     // ALPHA = 0.5 -> sqrt
    part = ws[WS_SUM + t] * w;                  // empty bin: sum==0 -> contributes 0
  }
  #pragma unroll
  for (int off = 16; off > 0; off >>= 1)
    part += __shfl_down(part, off, 32);
  if (t == 0) out[0] = part / total;
}

extern "C" void kernel_launch(void* const* d_in, const int* in_sizes, int n_in,
                              void* d_out, int out_size, void* d_ws, size_t ws_size,
                              hipStream_t stream) {
  const float* pred   = (const float*)d_in[0];
  const float* target = (const float*)d_in[1];
  float* ws  = (float*)d_ws;
  float* out = (float*)d_out;
  const int n  = in_sizes[0];
  const int n4 = n / 4;

  rangeloss_zero_ws<<<1, 64, 0, stream>>>(ws);

  int blocks = (n4 + 255) / 256;
  if (blocks > 2048) blocks = 2048;       // 524288 threads grid-stride; saturates HBM
  if (blocks < 1) blocks = 1;
  rangeloss_hist<<<blocks, 256, 0, stream>>>(pred, target, ws, n4, n);

  rangeloss_finalize<<<1, 32, 0, stream>>>(ws, out);
}